// DepthLSSTransform_ours_depth_quan_75033078661476
// MI455X (gfx1250) — compile-verified
//
#include <hip/hip_runtime.h>
#include <cstdint>
#include <cstddef>

// ---------------- types ----------------
typedef __attribute__((ext_vector_type(2))) float v2f;
typedef __attribute__((ext_vector_type(8))) float v8f;

#define NIMG 2
#define FH_ 32
#define FW_ 88
#define NPIX_CAM (FH_*FW_)   // 2816
#define DD 118
#define OUTCH 80
#define IMGH 256
#define IMGW 704
#define NXY 360
#define BEVPIX (NXY*NXY)     // 129600

// ---------------- 3x3 inverse of the top-left of a row-major 4x4 ----------------
__device__ inline void inv3x3(const float* m, float* o) {
  float a=m[0],b=m[1],c=m[2];
  float d=m[4],e=m[5],f=m[6];
  float g=m[8],h=m[9],i=m[10];
  float A=e*i-f*h, Bc=c*h-b*i, Cc=b*f-c*e;
  float Dd=f*g-d*i, Ee=a*i-c*g, Ff=c*d-a*f;
  float Gg=d*h-e*g, Hh=b*g-a*h, Ii=a*e-b*d;
  float det = a*A + b*Dd + c*Gg;
  float r = 1.0f/det;
  o[0]=A*r; o[1]=Bc*r; o[2]=Cc*r;
  o[3]=Dd*r; o[4]=Ee*r; o[5]=Ff*r;
  o[6]=Gg*r; o[7]=Hh*r; o[8]=Ii*r;
}

// ---------------- point-cloud -> depth image scatter ----------------
__global__ void k_scatter_depth(const float* __restrict__ pts,
                                const float* __restrict__ laug,
                                const float* __restrict__ l2i,
                                const float* __restrict__ iaug,
                                float* __restrict__ depth, int P) {
  int p = blockIdx.x*256 + threadIdx.x;
  if (p >= P) return;
  const float* q = pts + (size_t)p*4;
  int b = (int)q[0];
  const float* La = laug + b*16;
  float x = q[1] - La[3], y = q[2] - La[7], z = q[3] - La[11];
  float R[9]; inv3x3(La, R);
  float cx = R[0]*x + R[1]*y + R[2]*z;
  float cy = R[3]*x + R[4]*y + R[5]*z;
  float cz = R[6]*x + R[7]*y + R[8]*z;
  const float* L = l2i + b*16;
  float tx = L[0]*cx + L[1]*cy + L[2]*cz + L[3];
  float ty = L[4]*cx + L[5]*cy + L[6]*cz + L[7];
  float tz = L[8]*cx + L[9]*cy + L[10]*cz + L[11];
  float dist = tz;
  float zc = fminf(fmaxf(tz, 1e-5f), 1e5f);
  float u = tx/zc, v = ty/zc;
  const float* A = iaug + b*16;
  float uu = A[0]*u + A[1]*v + A[2]*zc + A[3];
  float vv = A[4]*u + A[5]*v + A[6]*zc + A[7];
  if (vv >= 0.f && vv < (float)IMGH && uu >= 0.f && uu < (float)IMGW) {
    int vi = (int)vv, ui = (int)uu;
    depth[((size_t)b*IMGH + vi)*IMGW + ui] = dist;
  }
}

// ---------------- small direct conv (dtransform layers) ----------------
struct SArgs {
  const float *in, *w, *bias, *gamma, *beta, *mean, *var;
  float* out;
  int Cin, Cout, IH, IW, OH, OW, K, stride, pad;
  long long tot;
};
__global__ __launch_bounds__(256) void k_sconv(SArgs a) {
  long long i = (long long)blockIdx.x*256 + threadIdx.x;
  if (i >= a.tot) return;
  int ox = (int)(i % a.OW);
  int oy = (int)((i / a.OW) % a.OH);
  int co = (int)((i / ((long long)a.OW*a.OH)) % a.Cout);
  int img = (int)(i / ((long long)a.OW*a.OH*a.Cout));
  float acc = a.bias[co];
  for (int ci = 0; ci < a.Cin; ++ci) {
    const float* ib = a.in + ((size_t)img*a.Cin + ci)*a.IH*a.IW;
    const float* wb = a.w + ((size_t)co*a.Cin + ci)*a.K*a.K;
    for (int ky = 0; ky < a.K; ++ky) {
      int iy = oy*a.stride + ky - a.pad;
      if (iy < 0 || iy >= a.IH) continue;
      for (int kx = 0; kx < a.K; ++kx) {
        int ix = ox*a.stride + kx - a.pad;
        if (ix < 0 || ix >= a.IW) continue;
        acc += ib[(size_t)iy*a.IW + ix] * wb[ky*a.K + kx];
      }
    }
  }
  float inv = a.gamma[co] * rsqrtf(a.var[co] + 1e-5f);
  float val = acc*inv + a.beta[co] - a.mean[co]*inv;
  a.out[i] = fmaxf(val, 0.f);
}

// ---------------- concat depth-features + fpn into f32 HWC ----------------
__global__ void k_concat(const float* __restrict__ d3, const float* __restrict__ fpn,
                         float* __restrict__ xh) {
  long long i = (long long)blockIdx.x*256 + threadIdx.x;
  long long tot = (long long)NIMG*NPIX_CAM*320;
  if (i >= tot) return;
  int c = (int)(i % 320);
  long long r = i / 320;
  int p = (int)(r % NPIX_CAM);
  int img = (int)(r / NPIX_CAM);
  xh[i] = (c < 64) ? d3[((size_t)img*64 + c)*NPIX_CAM + p]
                   : fpn[((size_t)img*256 + (c-64))*NPIX_CAM + p];
}

// ---------------- weight packer -> per-lane f32 WMMA A-fragment order ----------------
// A 16x4 f32 layout: lane (h=lane>>4, m=lane&15) holds row m, VGPR v -> K = h*2+v.
// We store, per 32-wide K chunk kc and co-tile ct:  [lane][ks(0..7)][v(0..1)] floats
// where k_local = ks*4 + h*2 + v.
struct PackArgs {
  const float* w; float* out;
  int WD1, S, swap_oc;   // value(co,ci,slot) = swap? w[(ci*WD1+co)*S+slot] : w[(co*WD1+ci)*S+slot]
  int ntap; int wslot[9];
  int Cin, Cout, CoTiles, cpg, KC;
};
__global__ __launch_bounds__(256) void k_pack(PackArgs a) {
  long long i = (long long)blockIdx.x*256 + threadIdx.x;
  long long tot = (long long)a.KC * a.CoTiles * 512;
  if (i >= tot) return;
  int v  = (int)(i & 1);
  int ks = (int)((i >> 1) & 7);
  int lane = (int)((i >> 4) & 31);
  long long rest = i >> 9;
  int ct = (int)(rest % a.CoTiles);
  int kc = (int)(rest / a.CoTiles);
  int hh = lane >> 4, m = lane & 15;
  int kl = ks*4 + hh*2 + v;
  int tap = kc / a.cpg;
  int cin = (kc - tap*a.cpg)*32 + kl;
  int co = ct*16 + m;
  float val = 0.f;
  if (co < a.Cout && cin < a.Cin) {
    int r0 = a.swap_oc ? cin : co;
    int c0 = a.swap_oc ? co  : cin;
    val = a.w[((size_t)r0*a.WD1 + c0)*a.S + a.wslot[tap]];
  }
  a.out[i] = val;
}

// ---------------- generic implicit-GEMM conv via v_wmma_f32_16x16x4_f32 ----------------
// input:  f32 HWC (channel-padded to Cpad, multiple of 32)  OR  f32 CHW (Cin_chw>0)
// output: optional f32 CHW and/or f32 HWC (padded to CoutPad, pad channels zeroed)
struct ConvArgs {
  const float* in; const float* wpack;
  float* out_chw; float* out_hwc;
  const float *bias, *gamma, *beta, *mean, *var;
  int IH, IW, Cpad, Cin_chw;     // Cin_chw>0 => input CHW with that many real channels
  int OH, OW, stride;            // virtual output grid for tiling / tap addressing
  int ntap; int dy[9]; int dx[9];
  int KC, cpg;
  int CoTiles, Cout, CoutPad;
  int OHr, OWr, omul, ooy, oox;  // real output coords: (oy*omul+ooy, ox*omul+oox)
  int relu;
};
__global__ __launch_bounds__(256) void k_wmma_conv(ConvArgs a) {
  __shared__ float lsB[64*36];   // 64 px rows x 32 k (stride 36 floats)
  int tid = threadIdx.x, wave = tid >> 5, lane = tid & 31;
  int img = blockIdx.z;
  int pxbase = blockIdx.x*64;
  int cot = blockIdx.y*8 + wave;
  int npx = a.OH * a.OW;

  v8f zero = {0.f,0.f,0.f,0.f,0.f,0.f,0.f,0.f};
  v8f acc[4]; acc[0]=zero; acc[1]=zero; acc[2]=zero; acc[3]=zero;

  int lp = pxbase + (tid >> 2);     // loader pixel for B tile
  int lq = tid & 3;
  int loy = lp / a.OW, lox = lp % a.OW;
  bool lpok = lp < npx;
  int hh = lane >> 4, nr = lane & 15;
  int act = __builtin_amdgcn_readfirstlane((cot < a.CoTiles) ? 1 : 0);
  const float* inb = (a.Cin_chw > 0)
      ? a.in + (size_t)img * a.Cin_chw * a.IH * a.IW
      : a.in + (size_t)img * a.IH * a.IW * a.Cpad;

  for (int kc = 0; kc < a.KC; ++kc) {
    int tap = kc / a.cpg;
    int c0 = (kc - tap*a.cpg)*32;
    int iy = loy*a.stride + a.dy[tap];
    int ix = lox*a.stride + a.dx[tap];
    bool ok = lpok && iy >= 0 && iy < a.IH && ix >= 0 && ix < a.IW;
    float* dst = &lsB[(size_t)(tid >> 2)*36 + lq*8];
    if (a.Cin_chw > 0) {
      #pragma unroll
      for (int j = 0; j < 8; ++j) {
        int c = c0 + lq*8 + j;
        dst[j] = (ok && c < a.Cin_chw) ? inb[((size_t)c*a.IH + iy)*a.IW + ix] : 0.f;
      }
    } else {
      uint4 v0 = make_uint4(0,0,0,0), v1 = v0;
      if (ok) {
        const float* s = inb + ((size_t)iy*a.IW + ix)*a.Cpad + c0 + lq*8;
        v0 = *(const uint4*)s; v1 = *(const uint4*)(s + 4);
      }
      *(uint4*)dst = v0; *(uint4*)(dst + 4) = v1;
    }
    __syncthreads();

    if (act) {
      // A fragments for all 8 K-substeps: 16 contiguous floats per lane
      const float* wp = a.wpack + ((size_t)kc*a.CoTiles + cot)*512 + lane*16;
      float av[16];
      #pragma unroll
      for (int t = 0; t < 16; ++t) av[t] = wp[t];
      #pragma unroll
      for (int ks = 0; ks < 8; ++ks) {
        v2f af; af.x = av[ks*2]; af.y = av[ks*2+1];
        #pragma unroll
        for (int j = 0; j < 4; ++j) {
          const float* row = &lsB[(size_t)(j*16 + nr)*36 + ks*4 + hh*2];
          v2f bf; bf.x = row[0]; bf.y = row[1];
          acc[j] = __builtin_amdgcn_wmma_f32_16x16x4_f32(
              false, af, false, bf, (short)0, acc[j], false, false);
        }
      }
    }
    __syncthreads();
  }

  if (!act) return;
  size_t opix = (size_t)a.OHr * a.OWr;
  #pragma unroll
  for (int r = 0; r < 8; ++r) {
    int co = cot*16 + hh*8 + r;
    float s = 1.f, t = 0.f;
    if (co < a.Cout) {
      float b = a.bias ? a.bias[co] : 0.f;
      if (a.gamma) {
        float inv = a.gamma[co] * rsqrtf(a.var[co] + 1e-5f);
        s = inv; t = b*inv + a.beta[co] - a.mean[co]*inv;
      } else t = b;
    }
    for (int j = 0; j < 4; ++j) {
      int p = pxbase + j*16 + nr;
      if (p >= npx) continue;
      float val = acc[j][r]*s + t;
      if (a.relu) val = fmaxf(val, 0.f);
      int ry = (p / a.OW)*a.omul + a.ooy;
      int rx = (p % a.OW)*a.omul + a.oox;
      size_t rp = (size_t)ry*a.OWr + rx;
      if (a.out_chw && co < a.Cout)
        a.out_chw[((size_t)img*a.Cout + co)*opix + rp] = val;
      if (a.out_hwc && co < a.CoutPad)
        a.out_hwc[((size_t)img*opix + rp)*a.CoutPad + co] = (co < a.Cout) ? val : 0.f;
    }
  }
}

// ---------------- channel softmax (CHW), works in-place ----------------
__global__ void k_softmax_ch(const float* __restrict__ in, float* __restrict__ out,
                             int C, int npx, int nimg) {
  int i = blockIdx.x*256 + threadIdx.x;
  if (i >= nimg*npx) return;
  int img = i / npx, p = i % npx;
  const float* src = in + (size_t)img*C*npx + p;
  float* dst = out + (size_t)img*C*npx + p;
  float m = -3.4e38f;
  for (int c = 0; c < C; ++c) m = fmaxf(m, src[(size_t)c*npx]);
  float s = 0.f;
  for (int c = 0; c < C; ++c) s += __expf(src[(size_t)c*npx] - m);
  float inv = 1.f/s;
  for (int c = 0; c < C; ++c) dst[(size_t)c*npx] = __expf(src[(size_t)c*npx] - m)*inv;
}

// ---------------- lift (softmax_depth x feature) + voxel scatter-add ----------------
__global__ void k_lift(const float* __restrict__ dsm, const float* __restrict__ feat,
                       const float* __restrict__ iaug, const float* __restrict__ intr,
                       const float* __restrict__ c2l, const float* __restrict__ laug,
                       float* __restrict__ grid) {
  int i = blockIdx.x*256 + threadIdx.x;
  const int TOT = NIMG*DD*NPIX_CAM;
  if (i >= TOT) return;
  int p = i % NPIX_CAM;
  int d = (i / NPIX_CAM) % DD;
  int img = i / (NPIX_CAM*DD);
  int hc = p / FW_, wc = p % FW_;
  float ds = 1.0f + 0.5f*(float)d;
  float xs = (float)wc * (703.0f/87.0f);
  float ys = (float)hc * (255.0f/31.0f);
  const float* A = iaug + img*16;
  float v0 = xs - A[3], v1 = ys - A[7], v2 = ds - A[11];
  float Ri[9]; inv3x3(A, Ri);
  float w0 = Ri[0]*v0 + Ri[1]*v1 + Ri[2]*v2;
  float w1 = Ri[3]*v0 + Ri[4]*v1 + Ri[5]*v2;
  float w2 = Ri[6]*v0 + Ri[7]*v1 + Ri[8]*v2;
  w0 *= w2; w1 *= w2;
  float Ki[9]; inv3x3(intr + img*16, Ki);
  float t0 = Ki[0]*w0 + Ki[1]*w1 + Ki[2]*w2;
  float t1 = Ki[3]*w0 + Ki[4]*w1 + Ki[5]*w2;
  float t2 = Ki[6]*w0 + Ki[7]*w1 + Ki[8]*w2;
  const float* C = c2l + img*16;
  float p0 = C[0]*t0 + C[1]*t1 + C[2]*t2 + C[3];
  float p1 = C[4]*t0 + C[5]*t1 + C[6]*t2 + C[7];
  float p2 = C[8]*t0 + C[9]*t1 + C[10]*t2 + C[11];
  const float* Lg = laug + img*16;   // b = img (N==1)
  float g0 = Lg[0]*p0 + Lg[1]*p1 + Lg[2]*p2 + Lg[3];
  float g1 = Lg[4]*p0 + Lg[5]*p1 + Lg[6]*p2 + Lg[7];
  float g2 = Lg[8]*p0 + Lg[9]*p1 + Lg[10]*p2 + Lg[11];
  int gx = (int)((g0 + 54.0f)/0.3f);
  int gy = (int)((g1 + 54.0f)/0.3f);
  int gz = (int)((g2 + 10.0f)/20.0f);
  if (gx < 0 || gx >= NXY || gy < 0 || gy >= NXY || gz != 0) return;
  float dv = dsm[((size_t)img*DD + d)*NPIX_CAM + p];
  float* gb = grid + ((size_t)img*OUTCH)*BEVPIX + (size_t)gx*NXY + gy;
  const float* fb = feat + (size_t)img*OUTCH*NPIX_CAM + p;
  for (int c = 0; c < OUTCH; ++c)
    atomicAdd(gb + (size_t)c*BEVPIX, dv * fb[(size_t)c*NPIX_CAM]);
}

// =======================================================================
//                               HOST SIDE
// =======================================================================
static inline size_t alignup(size_t x) { return (x + 255) & ~(size_t)255; }

static void conv_launch(hipStream_t st, const ConvArgs& a) {
  int npx = a.OH*a.OW;
  dim3 g((npx + 63)/64, (a.CoTiles + 7)/8, NIMG);
  k_wmma_conv<<<g, dim3(256), 0, st>>>(a);
}
static void pack_launch(hipStream_t st, const PackArgs& a) {
  long long tot = (long long)a.KC*a.CoTiles*512;
  k_pack<<<dim3((unsigned)((tot + 255)/256)), dim3(256), 0, st>>>(a);
}

extern "C" void kernel_launch(void* const* d_in, const int* in_sizes, int n_in,
                              void* d_out, int out_size, void* d_ws, size_t ws_size,
                              hipStream_t stream) {
  (void)in_sizes; (void)n_in; (void)out_size; (void)ws_size;
  auto IN = [&](int i) { return (const float*)d_in[i]; };
  // input index table (setup_inputs insertion order, params flattened in dict order)
  const float* fpn   = IN(0);
  const float* intr  = IN(1);
  const float* c2l   = IN(2);
  const float* iaug  = IN(3);
  const float* laug  = IN(4);
  const float* l2i   = IN(5);
  const float* pts   = IN(6);
  // dtransform: base 7 (3 convs x {w,b,gamma,beta,mean,var})
  // depthnet: base 25; features_out: base 43
  // upsample: 61..75 (3 dconvs x {w,gamma,beta,mean,var}), 76 w, 77 b
  // downsample: base 78
  char* ws = (char*)d_ws;
  size_t off = 0;
  auto A = [&](size_t bytes) { off = alignup(off); size_t o = off; off += bytes; return o; };

  const size_t IPIX = (size_t)IMGH*IMGW;           // 180224
  // --- persistent buffers (live across phases) ---
  size_t o_dl    = A((size_t)NIMG*DD*NPIX_CAM*4);
  size_t o_dlf   = A((size_t)NIMG*NPIX_CAM*128*4);   // dl in HWC(128) for deconv input
  size_t o_feat  = A((size_t)NIMG*OUTCH*NPIX_CAM*4);
  size_t o_dsm   = A((size_t)NIMG*DD*NPIX_CAM*4);
  // weight packs (f32)
  size_t o_pDN0 = A((size_t)90*16*512*4);
  size_t o_pDN1 = A((size_t)72*16*512*4);
  size_t o_pDN2 = A((size_t)8*8*512*4);
  size_t o_pF0  = A((size_t)90*16*512*4);
  size_t o_pF1  = A((size_t)72*16*512*4);
  size_t o_pF2  = A((size_t)8*5*512*4);
  size_t o_pUP  = A((size_t)4*8*512*4);
  size_t o_pDC[3][4];
  const int decNTAP[4] = {9, 6, 6, 4};  // (ey,ex) = (0,0),(0,1),(1,0),(1,1)
  for (int dci = 0; dci < 3; ++dci)
    for (int cl = 0; cl < 4; ++cl)
      o_pDC[dci][cl] = A((size_t)(decNTAP[cl]*4)*8*512*4);
  size_t o_pDS[3];
  for (int j = 0; j < 3; ++j) o_pDS[j] = A((size_t)27*5*512*4);

  // --- phase-overlaid union region ---
  size_t ub = alignup(off);
  size_t uo = ub;
  auto U = [&](size_t bytes) { uo = alignup(uo); size_t o = uo; uo += bytes; return o; };
  // phase 1: depth/dtransform/concat/depthnet/features
  size_t o_depth = U(NIMG*IPIX*4);
  size_t o_d1    = U((size_t)NIMG*8*IPIX*4);
  size_t o_d2    = U((size_t)NIMG*32*64*176*4);
  size_t o_d3    = U((size_t)NIMG*64*NPIX_CAM*4);
  size_t o_xh    = U((size_t)NIMG*NPIX_CAM*320*4);
  size_t o_yA    = U((size_t)NIMG*NPIX_CAM*256*4);
  size_t o_yB    = U((size_t)NIMG*NPIX_CAM*256*4);
  size_t e1 = uo;
  // phase 2: deconv pyramid (u3 computed per parity class, compact)
  uo = ub;
  size_t o_u1  = U((size_t)NIMG*64*176*128*4);
  size_t o_u2  = U((size_t)NIMG*128*352*128*4);
  size_t o_u3c = U((size_t)NIMG*128*352*128*4);
  size_t e2 = uo;
  // phase 3: BEV grid (CHW, atomics) + downsample activations (HWC 96)
  uo = ub;
  size_t o_grid = U((size_t)NIMG*OUTCH*BEVPIX*4);
  size_t o_dsA  = U((size_t)NIMG*BEVPIX*96*4);
  size_t o_dsB  = U((size_t)NIMG*180*180*96*4);
  size_t e3 = uo;
  off = e1 > e2 ? e1 : e2; if (e3 > off) off = e3;

  float* dout = (float*)d_out;
  float* out_bev = dout;                                   // (2,80,180,180)
  float* out_dp  = dout + (size_t)NIMG*OUTCH*180*180;      // (2,118,256,704)

  // ---------------- 1) depth scatter ----------------
  hipMemsetAsync(ws + o_depth, 0, NIMG*IPIX*4, stream);
  k_scatter_depth<<<dim3((200000 + 255)/256), dim3(256), 0, stream>>>(
      pts, laug, l2i, iaug, (float*)(ws + o_depth), 200000);

  // ---------------- 2) dtransform (scalar convs) ----------------
  {
    SArgs s{}; s.in=(float*)(ws+o_depth); s.w=IN(7); s.bias=IN(8); s.gamma=IN(9); s.beta=IN(10); s.mean=IN(11); s.var=IN(12);
    s.out=(float*)(ws+o_d1); s.Cin=1; s.Cout=8; s.IH=IMGH; s.IW=IMGW; s.OH=IMGH; s.OW=IMGW; s.K=1; s.stride=1; s.pad=0;
    s.tot=(long long)NIMG*8*IPIX;
    k_sconv<<<dim3((unsigned)((s.tot+255)/256)), dim3(256), 0, stream>>>(s);
  }
  {
    SArgs s{}; s.in=(float*)(ws+o_d1); s.w=IN(13); s.bias=IN(14); s.gamma=IN(15); s.beta=IN(16); s.mean=IN(17); s.var=IN(18);
    s.out=(float*)(ws+o_d2); s.Cin=8; s.Cout=32; s.IH=IMGH; s.IW=IMGW; s.OH=64; s.OW=176; s.K=5; s.stride=4; s.pad=2;
    s.tot=(long long)NIMG*32*64*176;
    k_sconv<<<dim3((unsigned)((s.tot+255)/256)), dim3(256), 0, stream>>>(s);
  }
  {
    SArgs s{}; s.in=(float*)(ws+o_d2); s.w=IN(19); s.bias=IN(20); s.gamma=IN(21); s.beta=IN(22); s.mean=IN(23); s.var=IN(24);
    s.out=(float*)(ws+o_d3); s.Cin=32; s.Cout=64; s.IH=64; s.IW=176; s.OH=FH_; s.OW=FW_; s.K=5; s.stride=2; s.pad=2;
    s.tot=(long long)NIMG*64*NPIX_CAM;
    k_sconv<<<dim3((unsigned)((s.tot+255)/256)), dim3(256), 0, stream>>>(s);
  }
  {
    long long tot = (long long)NIMG*NPIX_CAM*320;
    k_concat<<<dim3((unsigned)((tot+255)/256)), dim3(256), 0, stream>>>(
        (float*)(ws+o_d3), fpn, (float*)(ws+o_xh));
  }

  // ---------------- 3) weight packs ----------------
  auto mkpack = [&](const float* w, size_t po, int WD1, int S, int swap_oc,
                    int K, int Cpad, int Cin, int Cout) {
    PackArgs pa{}; pa.w=w; pa.out=(float*)(ws+po); pa.WD1=WD1; pa.S=S; pa.swap_oc=swap_oc;
    pa.ntap=K*K;
    for (int ky=0;ky<K;++ky) for (int kx=0;kx<K;++kx) pa.wslot[ky*K+kx]=ky*K+kx;
    pa.Cin=Cin; pa.Cout=Cout; pa.CoTiles=(Cout+15)/16; pa.cpg=Cpad/32; pa.KC=pa.ntap*pa.cpg;
    pack_launch(stream, pa);
  };
  mkpack(IN(25), o_pDN0, 320, 9, 0, 3, 320, 320, 256);
  mkpack(IN(31), o_pDN1, 256, 9, 0, 3, 256, 256, 256);
  mkpack(IN(37), o_pDN2, 256, 1, 0, 1, 256, 256, 118);
  mkpack(IN(43), o_pF0,  320, 9, 0, 3, 320, 320, 256);
  mkpack(IN(49), o_pF1,  256, 9, 0, 3, 256, 256, 256);
  mkpack(IN(55), o_pF2,  256, 1, 0, 1, 256, 256, 80);
  mkpack(IN(76), o_pUP,  118, 1, 0, 1, 128, 118, 118);
  // deconv class packs: w'[co,ci,ky,kx] = w[ci,co,4-ky,4-kx]
  int dcdy[4][9], dcdx[4][9], dcn[4];
  for (int cl = 0; cl < 4; ++cl) {
    int ey = cl >> 1, ex = cl & 1, n = 0;
    int kys[3], nky = 0, kxs[3], nkx = 0;
    for (int k = 0; k < 5; ++k) { if (((ey + k) & 1) == 0) kys[nky++] = k; }
    for (int k = 0; k < 5; ++k) { if (((ex + k) & 1) == 0) kxs[nkx++] = k; }
    for (int a0 = 0; a0 < nky; ++a0) for (int b0 = 0; b0 < nkx; ++b0) {
      dcdy[cl][n] = (ey + kys[a0] - 2)/2;
      dcdx[cl][n] = (ex + kxs[b0] - 2)/2;
      ++n;
    }
    dcn[cl] = n;
    for (int dci = 0; dci < 3; ++dci) {
      PackArgs pa{}; pa.w = IN(61 + 5*dci); pa.out=(float*)(ws+o_pDC[dci][cl]);
      pa.WD1=118; pa.S=25; pa.swap_oc=1; pa.ntap=n;
      int m = 0;
      for (int a0 = 0; a0 < nky; ++a0) for (int b0 = 0; b0 < nkx; ++b0)
        pa.wslot[m++] = (4 - kys[a0])*5 + (4 - kxs[b0]);
      pa.Cin=118; pa.Cout=118; pa.CoTiles=8; pa.cpg=4; pa.KC=n*4;
      pack_launch(stream, pa);
    }
  }
  for (int j = 0; j < 3; ++j)
    mkpack(IN(78 + 6*j), o_pDS[j], 80, 9, 0, 3, 96, 80, 80);

  // ---------------- 4) depthnet & features (WMMA f32) ----------------
  auto mkconv3 = [&](const float* in, size_t po, float* ochw, float* ohwc,
                     int ibase, int IH, int IW, int Cpad, int Cout, int CoutPad,
                     int K, int relu) {
    ConvArgs c{}; c.in=in; c.wpack=(float*)(ws+po); c.out_chw=ochw; c.out_hwc=ohwc;
    c.bias=IN(ibase+1); c.gamma=IN(ibase+2); c.beta=IN(ibase+3); c.mean=IN(ibase+4); c.var=IN(ibase+5);
    c.IH=IH; c.IW=IW; c.Cpad=Cpad; c.Cin_chw=0; c.OH=IH; c.OW=IW; c.stride=1;
    c.ntap=K*K; int pad=(K==3)?1:0;
    for (int ky=0;ky<K;++ky) for (int kx=0;kx<K;++kx){ c.dy[ky*K+kx]=ky-pad; c.dx[ky*K+kx]=kx-pad; }
    c.cpg=Cpad/32; c.KC=c.ntap*c.cpg; c.CoTiles=(Cout+15)/16; c.Cout=Cout; c.CoutPad=CoutPad;
    c.OHr=IH; c.OWr=IW; c.omul=1; c.ooy=0; c.oox=0; c.relu=relu;
    conv_launch(stream, c);
  };
  mkconv3((float*)(ws+o_xh), o_pDN0, nullptr, (float*)(ws+o_yA), 25, FH_, FW_, 320, 256, 256, 3, 1);
  mkconv3((float*)(ws+o_yA), o_pDN1, nullptr, (float*)(ws+o_yB), 31, FH_, FW_, 256, 256, 256, 3, 1);
  mkconv3((float*)(ws+o_yB), o_pDN2, (float*)(ws+o_dl), (float*)(ws+o_dlf), 37, FH_, FW_, 256, 118, 128, 1, 1);
  mkconv3((float*)(ws+o_xh), o_pF0, nullptr, (float*)(ws+o_yA), 43, FH_, FW_, 320, 256, 256, 3, 1);
  mkconv3((float*)(ws+o_yA), o_pF1, nullptr, (float*)(ws+o_yB), 49, FH_, FW_, 256, 256, 256, 3, 1);
  mkconv3((float*)(ws+o_yB), o_pF2, (float*)(ws+o_feat), nullptr, 55, FH_, FW_, 256, 80, 0, 1, 1);

  // dsm = softmax(dl)
  k_softmax_ch<<<dim3((NIMG*NPIX_CAM + 255)/256), dim3(256), 0, stream>>>(
      (float*)(ws+o_dl), (float*)(ws+o_dsm), DD, NPIX_CAM, NIMG);

  // ---------------- 5) deconv pyramid ----------------
  const size_t o_uin[2]  = { o_dlf, o_u1 };
  const size_t o_uout[2] = { o_u1, o_u2 };
  const int uIH[3] = { FH_, 64, 128 }, uIW[3] = { FW_, 176, 352 };
  auto mkdeconv = [&](int dci, int cl, const float* in, float* ohwc,
                      int omul, int ooy, int oox, int OHr, int OWr) {
    ConvArgs c{}; c.in=in; c.wpack=(float*)(ws+o_pDC[dci][cl]);
    c.out_chw=nullptr; c.out_hwc=ohwc;
    int ib = 61 + 5*dci;
    c.gamma=IN(ib+1); c.beta=IN(ib+2); c.mean=IN(ib+3); c.var=IN(ib+4);
    c.IH=uIH[dci]; c.IW=uIW[dci]; c.Cpad=128; c.Cin_chw=0;
    c.OH=uIH[dci]; c.OW=uIW[dci]; c.stride=1;
    c.ntap=dcn[cl];
    for (int t = 0; t < dcn[cl]; ++t) { c.dy[t]=dcdy[cl][t]; c.dx[t]=dcdx[cl][t]; }
    c.cpg=4; c.KC=c.ntap*4; c.CoTiles=8; c.Cout=118; c.CoutPad=128;
    c.OHr=OHr; c.OWr=OWr; c.omul=omul; c.ooy=ooy; c.oox=oox; c.relu=1;
    conv_launch(stream, c);
  };
  // deconv 1 and 2: write real strided positions into dense next-level HWC buffers
  for (int dci = 0; dci < 2; ++dci)
    for (int cl = 0; cl < 4; ++cl)
      mkdeconv(dci, cl, (float*)(ws+o_uin[dci]), (float*)(ws+o_uout[dci]),
               2, cl>>1, cl&1, 2*uIH[dci], 2*uIW[dci]);
  // deconv 3: per class -> compact u3c, then 1x1 conv scatters logits into d_out
  for (int cl = 0; cl < 4; ++cl) {
    mkdeconv(2, cl, (float*)(ws+o_u2), (float*)(ws+o_u3c), 1, 0, 0, 128, 352);
    ConvArgs c{}; c.in=(float*)(ws+o_u3c); c.wpack=(float*)(ws+o_pUP);
    c.out_chw=out_dp; c.out_hwc=nullptr; c.bias=IN(77);
    c.IH=128; c.IW=352; c.Cpad=128; c.Cin_chw=0; c.OH=128; c.OW=352; c.stride=1;
    c.ntap=1; c.dy[0]=0; c.dx[0]=0; c.cpg=4; c.KC=4; c.CoTiles=8; c.Cout=118; c.CoutPad=0;
    c.OHr=IMGH; c.OWr=IMGW; c.omul=2; c.ooy=cl>>1; c.oox=cl&1; c.relu=0;
    conv_launch(stream, c);
  }
  // softmax in place -> depth_pre
  k_softmax_ch<<<dim3((int)((NIMG*IPIX + 255)/256)), dim3(256), 0, stream>>>(
      out_dp, out_dp, DD, (int)IPIX, NIMG);

  // ---------------- 6) lift + BEV scatter ----------------
  hipMemsetAsync(ws + o_grid, 0, (size_t)NIMG*OUTCH*BEVPIX*4, stream);
  k_lift<<<dim3((NIMG*DD*NPIX_CAM + 255)/256), dim3(256), 0, stream>>>(
      (float*)(ws+o_dsm), (float*)(ws+o_feat), iaug, intr, c2l, laug, (float*)(ws+o_grid));

  // ---------------- 7) BEV downsample convs (WMMA f32) ----------------
  hipMemsetAsync(ws + o_dsA, 0, (size_t)NIMG*BEVPIX*96*4, stream);
  hipMemsetAsync(ws + o_dsB, 0, (size_t)NIMG*180*180*96*4, stream);
  auto mkds = [&](const float* in, size_t po, float* ochw, float* ohwc, int ibase,
                  int IH, int IW, int stride, int OH, int OW, int Cin_chw) {
    ConvArgs c{}; c.in=in; c.wpack=(float*)(ws+po); c.out_chw=ochw; c.out_hwc=ohwc;
    c.bias=IN(ibase+1); c.gamma=IN(ibase+2); c.beta=IN(ibase+3); c.mean=IN(ibase+4); c.var=IN(ibase+5);
    c.IH=IH; c.IW=IW; c.Cpad=96; c.Cin_chw=Cin_chw; c.OH=OH; c.OW=OW; c.stride=stride;
    c.ntap=9;
    for (int ky=0;ky<3;++ky) for (int kx=0;kx<3;++kx){ c.dy[ky*3+kx]=ky-1; c.dx[ky*3+kx]=kx-1; }
    c.cpg=3; c.KC=27; c.CoTiles=5; c.Cout=80; c.CoutPad=ohwc?96:0;
    c.OHr=OH; c.OWr=OW; c.omul=1; c.ooy=0; c.oox=0; c.relu=1;
    conv_launch(stream, c);
  };
  mkds((float*)(ws+o_grid), o_pDS[0], nullptr, (float*)(ws+o_dsA), 78, NXY, NXY, 1, NXY, NXY, OUTCH);
  mkds((float*)(ws+o_dsA),  o_pDS[1], nullptr, (float*)(ws+o_dsB), 84, NXY, NXY, 2, 180, 180, 0);
  mkds((float*)(ws+o_dsB),  o_pDS[2], out_bev, nullptr,            90, 180, 180, 1, 180, 180, 0);
}